// Dilation1D_67199058313279
// MI455X (gfx1250) — compile-verified
//
#include <hip/hip_runtime.h>
#include <stdint.h>

// out[x] = max(0, max_{y=-20..20, 0<=x-y<N} input[x-y] - y^2/(4*scale))
//
// Strategy (MI455X / gfx1250):
//  - memory-bound-ish (32 MB traffic, ~1.4us floor @ 23.3 TB/s): read each
//    input element once via an LDS-staged tile with halo.
//  - staging uses the CDNA5 async global->LDS path (ASYNCcnt):
//        global_load_async_to_lds_b128  +  s_wait_asynccnt 0
//  - compute: sliding-window max-plus, 4 outputs/thread sharing loads,
//    41 bias values held in VGPRs after full unroll.

#define K_WIN   41
#define HALF    20
#define TPB     256
#define R_PER   4
#define TILE    (TPB * R_PER)          // 1024 outputs per block
#define PRE     32                     // left pad, keeps B128 chunks aligned
#define SMEM_FLOATS (PRE + TILE + 64)  // 1120 floats = 4480 B (covers halo)
#define SMEM_VEC4   (SMEM_FLOATS / 4)  // 280 float4 chunks

__global__ __launch_bounds__(TPB) void dilation1d_kernel(
    const float* __restrict__ in, const float* __restrict__ scale_p,
    float* __restrict__ out, int n) {
  __shared__ float smem[SMEM_FLOATS];

  const int tid   = threadIdx.x;
  const int tile0 = (int)blockIdx.x * TILE;

  // ---------- async global -> LDS staging (gfx1250-specific) ----------
  // Low 32 bits of a generic LDS pointer are the LDS byte offset.
  const uint32_t lds0  = (uint32_t)(uintptr_t)(&smem[0]);
  const uint64_t gbase = (uint64_t)(uintptr_t)in;

  {
    // round 1: chunks 0..255 (one float4 per lane)
    int c  = tid;
    int g4 = tile0 - PRE + 4 * c;                 // global element index
    g4 = g4 < 0 ? 0 : g4;                         // clamp (edge blocks load
    g4 = g4 > n - 4 ? n - 4 : g4;                 //  safe garbage; masked later)
    uint32_t goff  = (uint32_t)(g4 * 4);          // byte offset (fits 32b)
    uint32_t laddr = lds0 + (uint32_t)(c * 16);
    asm volatile("global_load_async_to_lds_b128 %0, %1, %2"
                 :: "v"(laddr), "v"(goff), "s"(gbase) : "memory");
  }
  if (tid < SMEM_VEC4 - TPB) {
    // round 2: chunks 256..279 (halo tail)
    int c  = TPB + tid;
    int g4 = tile0 - PRE + 4 * c;
    g4 = g4 < 0 ? 0 : g4;
    g4 = g4 > n - 4 ? n - 4 : g4;
    uint32_t goff  = (uint32_t)(g4 * 4);
    uint32_t laddr = lds0 + (uint32_t)(c * 16);
    asm volatile("global_load_async_to_lds_b128 %0, %1, %2"
                 :: "v"(laddr), "v"(goff), "s"(gbase) : "memory");
  }
  asm volatile("s_wait_asynccnt 0x0" ::: "memory");
  __syncthreads();

  // ---------- compute ----------
  const float sc    = scale_p[0];
  const float inv4s = 1.0f / (4.0f * sc);
  float h[K_WIN];
#pragma unroll
  for (int k = 0; k < K_WIN; ++k) {
    float y = (float)(k - HALF);
    h[k] = -(y * y) * inv4s;          // = -(20-k)^2/(4s), symmetric
  }

  const int   x0    = tile0 + tid * R_PER;
  const int   lbase = tid * R_PER + (PRE - HALF);   // 16B-aligned LDS base
  const float NEG   = -__builtin_inff();

  float acc[R_PER];
#pragma unroll
  for (int r = 0; r < R_PER; ++r) acc[r] = NEG;

  const bool interior = (tile0 - HALF >= 0) && (tile0 + TILE - 1 + HALF < n);

  if (interior) {
#pragma unroll
    for (int j = 0; j < R_PER + K_WIN - 1; ++j) {   // 44 shared LDS loads
      float v = smem[lbase + j];
#pragma unroll
      for (int r = 0; r < R_PER; ++r) {
        int k = j - r;                              // compile-time constant
        if (k >= 0 && k < K_WIN) acc[r] = fmaxf(acc[r], v + h[k]);
      }
    }
    float4 o;
    o.x = fmaxf(acc[0], 0.0f);
    o.y = fmaxf(acc[1], 0.0f);
    o.z = fmaxf(acc[2], 0.0f);
    o.w = fmaxf(acc[3], 0.0f);
    *(float4*)(out + x0) = o;                       // 16B-aligned store
  } else {
#pragma unroll
    for (int j = 0; j < R_PER + K_WIN - 1; ++j) {
      int   g = x0 - HALF + j;                      // true global tap index
      float v = (g >= 0 && g < n) ? smem[lbase + j] : NEG;
#pragma unroll
      for (int r = 0; r < R_PER; ++r) {
        int k = j - r;
        if (k >= 0 && k < K_WIN) acc[r] = fmaxf(acc[r], v + h[k]);
      }
    }
#pragma unroll
    for (int r = 0; r < R_PER; ++r)
      if (x0 + r < n) out[x0 + r] = fmaxf(acc[r], 0.0f);
  }
}

extern "C" void kernel_launch(void* const* d_in, const int* in_sizes, int n_in,
                              void* d_out, int out_size, void* d_ws, size_t ws_size,
                              hipStream_t stream) {
  const float* in    = (const float*)d_in[0];
  const float* scale = (const float*)d_in[1];
  float*       out   = (float*)d_out;
  const int    n     = in_sizes[0];

  const int grid = (n + TILE - 1) / TILE;
  dilation1d_kernel<<<grid, TPB, 0, stream>>>(in, scale, out, n);
}